// CortalMagnification_16355235463637
// MI455X (gfx1250) — compile-verified
//
#include <hip/hip_runtime.h>
#include <hip/hip_bf16.h>
#include <stdint.h>

// Problem constants (from reference)
#define HH    512
#define WW    512
#define BATCH 32
#define HWSZ  (HH*WW)           // 262144
#define FOVC  160.0f            // 0.3125 * 512
#define KC    106.496f          // 0.208 * 512

// Reduction geometry
#define RB_TILES 32             // blocks per batch
#define RB_PIX   8192           // pixels per block (16 rows)
#define CHUNK    2048           // floats per TDM stage (8 KB)
#define NSTAGE   (RB_PIX/CHUNK) // 4

typedef float v2f __attribute__((ext_vector_type(2)));
typedef float v8f __attribute__((ext_vector_type(8)));
typedef unsigned int v4u __attribute__((ext_vector_type(4)));
typedef int v4i __attribute__((ext_vector_type(4)));
typedef int v8i __attribute__((ext_vector_type(8)));

#if __has_builtin(__builtin_amdgcn_wmma_f32_16x16x4_f32)
#define HAVE_WMMA4 1
#endif
#if __has_builtin(__builtin_amdgcn_tensor_load_to_lds) && __has_builtin(__builtin_amdgcn_s_wait_tensorcnt)
#define HAVE_TDM 1
#endif

// ---------------------------------------------------------------------------
// Wave32 sum via V_WMMA_F32_16X16X4_F32.
// A[m,k]: lane L<16 -> (k=0)=comp0,(k=1)=comp1 ; lane L>=16 -> (k=2),(k=3).
// Put value in comp0, 0 in comp1; B = ones(4x16).
// D[m,n] = p_m + p_{m+16}. Summing D's 8 VGPRs per lane gives partial row
// sums; lanes 0-15 hold sum over M=0..7, lanes 16-31 over M=8..15.
// total = t[lane0] + t[lane16].
// ---------------------------------------------------------------------------
__device__ __forceinline__ float wave_sum(float v) {
#ifdef HAVE_WMMA4
  v2f a; a[0] = v;    a[1] = 0.0f;
  v2f b; b[0] = 1.0f; b[1] = 1.0f;
  v8f c = {0.f,0.f,0.f,0.f,0.f,0.f,0.f,0.f};
  v8f d = __builtin_amdgcn_wmma_f32_16x16x4_f32(false, a, false, b,
                                                (short)0, c, false, false);
  float t = d[0]+d[1]+d[2]+d[3]+d[4]+d[5]+d[6]+d[7];
  return __shfl(t, 0, 32) + __shfl(t, 16, 32);
#else
  #pragma unroll
  for (int off = 16; off > 0; off >>= 1) v += __shfl_xor(v, off, 32);
  return v;
#endif
}

#ifdef HAVE_TDM
typedef __attribute__((address_space(3))) float lds_float;

// Issue one TDM descriptor: copy `nelem` contiguous f32 from gsrc -> LDS off.
// D# group0: count=1, lds_addr, 57-bit global_addr, type=2.
// D# group1: data_size=4B, 1-row tile (tile_dim0=nelem, tile_dim1=1).
__device__ __forceinline__ void tdm_load_f32(const float* gsrc,
                                             unsigned lds_byte_off,
                                             unsigned nelem) {
  unsigned long long ga = (unsigned long long)gsrc;
  v4u g0;
  g0[0] = 1u;                                   // count=1, flags 0
  g0[1] = lds_byte_off;                         // lds_addr [63:32]
  g0[2] = (unsigned)(ga & 0xFFFFFFFFull);       // global_addr lo
  g0[3] = (unsigned)((ga >> 32) & 0x01FFFFFFull) | (2u << 30); // addr hi + type=2
  const unsigned td0 = 1u << 20;                // tensor_dim0 (large, no OOB)
  int w0 = (int)(2u << 16);                     // data_size=2 (4B); mask=0
  int w1 = (int)((td0 & 0xFFFFu) << 16);        // tensor_dim0[15:0]
  int w2 = (int)((td0 >> 16) & 0xFFFFu) | (1 << 16); // td0[31:16] | tensor_dim1=1 lo
  int w3 = (int)((nelem & 0xFFFFu) << 16);      // tensor_dim1 hi=0 | tile_dim0
  int w4 = 1;                                   // tile_dim1=1, tile_dim2=0
  int w5 = (int)td0;                            // tensor_dim0_stride lo32
  int w6 = 0;                                   // stride hi / dim1_stride lo
  int w7 = 0;
  v8i g1 = { w0, w1, w2, w3, w4, w5, w6, w7 };
  v4i gz = { 0, 0, 0, 0 };
#if __clang_major__ >= 23
  v8i gz8 = { 0, 0, 0, 0, 0, 0, 0, 0 };
  __builtin_amdgcn_tensor_load_to_lds(g0, g1, gz, gz, gz8, 0);
#else
  __builtin_amdgcn_tensor_load_to_lds(g0, g1, gz, gz, 0);
#endif
}
#endif // HAVE_TDM

// ---------------------------------------------------------------------------
// Kernel A: per (batch, tile) partial sums of S, S*r, S*r^2.
// TDM double-buffers 8KB saliency tiles into LDS (wave 0 issues & waits).
// ---------------------------------------------------------------------------
__global__ __launch_bounds__(256)
void cm_reduce_kernel(const float* __restrict__ sal,
                      const float* __restrict__ gaze,
                      float* __restrict__ partials) {
  __shared__ float red[8][3];
  const int b   = blockIdx.y;
  const int t   = blockIdx.x;
  const int tid = threadIdx.x;
  const float xg = gaze[2*b + 0];
  const float yg = gaze[2*b + 1];
  const float* src = sal + (size_t)b * HWSZ + (size_t)t * RB_PIX;

  float aS = 0.f, aSr = 0.f, aSr2 = 0.f;

#ifdef HAVE_TDM
  __shared__ float tilebuf[2][CHUNK];
  const bool issuer = (tid < 32);               // wave 0 (uniform per wave)
  const unsigned lds0 =
      (unsigned)(unsigned long long)(lds_float*)&tilebuf[0][0];
  const unsigned lds1 =
      (unsigned)(unsigned long long)(lds_float*)&tilebuf[1][0];
  if (issuer) tdm_load_f32(src, lds0, CHUNK);
  for (int s = 0; s < NSTAGE; ++s) {
    if (issuer) {
      if (s + 1 < NSTAGE) {
        tdm_load_f32(src + (s + 1) * CHUNK, ((s + 1) & 1) ? lds1 : lds0, CHUNK);
        __builtin_amdgcn_s_wait_tensorcnt((short)1);   // stage s done (in-order)
      } else {
        __builtin_amdgcn_s_wait_tensorcnt((short)0);
      }
    }
    __syncthreads();
    const float* tb = tilebuf[s & 1];
    #pragma unroll
    for (int j = 0; j < CHUNK / 256; ++j) {
      const int e = tid + j * 256;
      const int p = t * RB_PIX + s * CHUNK + e;
      const float sv = tb[e];
      const float dx = (float)(p & (WW - 1)) - xg;
      const float dy = (float)(p >> 9) - yg;
      const float r = sqrtf(dx * dx + dy * dy + 1e-6f);
      aS += sv;
      aSr += sv * r;
      aSr2 += sv * r * r;
    }
    __syncthreads();   // buffer may be refilled two stages later
  }
#else
  for (int s = 0; s < NSTAGE; ++s) {
    #pragma unroll
    for (int j = 0; j < CHUNK / 256; ++j) {
      const int e = tid + j * 256;
      const int p = t * RB_PIX + s * CHUNK + e;
      const float sv = src[s * CHUNK + e];
      const float dx = (float)(p & (WW - 1)) - xg;
      const float dy = (float)(p >> 9) - yg;
      const float r = sqrtf(dx * dx + dy * dy + 1e-6f);
      aS += sv; aSr += sv * r; aSr2 += sv * r * r;
    }
  }
#endif

  // Cross-lane sums on the WMMA unit, then deterministic cross-wave sum.
  const float tS  = wave_sum(aS);
  const float tSr = wave_sum(aSr);
  const float tS2 = wave_sum(aSr2);
  const int wid  = tid >> 5;
  const int lane = tid & 31;
  if (lane == 0) { red[wid][0] = tS; red[wid][1] = tSr; red[wid][2] = tS2; }
  __syncthreads();
  if (tid == 0) {
    float s0 = 0.f, s1 = 0.f, s2 = 0.f;
    #pragma unroll
    for (int w = 0; w < 8; ++w) { s0 += red[w][0]; s1 += red[w][1]; s2 += red[w][2]; }
    float* o = partials + (size_t)(b * RB_TILES + t) * 3;
    o[0] = s0; o[1] = s1; o[2] = s2;
  }
}

// ---------------------------------------------------------------------------
// Kernel B: per-batch scalar parameters.
// r_max is attained at a corner (closed form); r_tfm is monotone in r, so
// max(r_tfm) = r_tfm(r_max) (closed form). Emits {xg, yg, fov_eff, coef}.
// ---------------------------------------------------------------------------
__global__ void cm_param_kernel(const float* __restrict__ partials,
                                const float* __restrict__ gaze,
                                float* __restrict__ params) {
  const int b = threadIdx.x;
  if (b >= BATCH) return;
  float sS = 0.f, sSr = 0.f, sSr2 = 0.f;
  for (int k = 0; k < RB_TILES; ++k) {
    const float* p = partials + (size_t)(b * RB_TILES + k) * 3;
    sS += p[0]; sSr += p[1]; sSr2 += p[2];
  }
  const float denom  = sS + 1e-6f;
  const float mean_r = sSr / denom;
  const float mean_r2 = sSr2 / denom;
  const float var = mean_r2 - mean_r * mean_r;
  const float stdr = sqrtf(fmaxf(var, 0.0f));
  const float xg = gaze[2*b], yg = gaze[2*b + 1];
  const float dxm = fmaxf(xg * xg, ((WW - 1) - xg) * ((WW - 1) - xg));
  const float dym = fmaxf(yg * yg, ((HH - 1) - yg) * ((HH - 1) - yg));
  const float rmax = sqrtf(dxm + dym + 1e-6f);
  const float spread = stdr / rmax;
  const float fov_eff = FOVC * (1.0f + 0.5f * spread);
  const float rpk = rmax + KC;
  const float rperi = rpk * rpk / (2.0f * (fov_eff + KC)) + (fov_eff - KC) * 0.5f;
  const float rtm = (rmax < fov_eff) ? rmax : rperi;
  const float coef = rmax / (rtm + 1e-6f);
  float4 pr; pr.x = xg; pr.y = yg; pr.z = fov_eff; pr.w = coef;
  ((float4*)params)[b] = pr;
}

// ---------------------------------------------------------------------------
// Kernel C: per-pixel radial warp + bilinear gather (zero border), clamp,
// uint8 truncation (stored as f32 per harness output convention).
// ---------------------------------------------------------------------------
__global__ __launch_bounds__(256)
void cm_warp_kernel(const float* __restrict__ img,
                    const float* __restrict__ params,
                    float* __restrict__ out) {
  const int b = blockIdx.y;
  const int p = blockIdx.x * 256 + threadIdx.x;
  const float4 pr = ((const float4*)params)[b];
  const float xg = pr.x, yg = pr.y, fov_eff = pr.z, coef = pr.w;

  const float dx = (float)(p & (WW - 1)) - xg;
  const float dy = (float)(p >> 9) - yg;
  const float r = sqrtf(dx * dx + dy * dy + 1e-6f);
  const float rpk = r + KC;
  const float r_peri = rpk * rpk / (2.0f * (fov_eff + KC)) + (fov_eff - KC) * 0.5f;
  const float r_tfm = (r < fov_eff) ? r : r_peri;
  const float r_new = coef * r_tfm;
  const float Xn = xg + (dx / r) * r_new;
  const float Yn = yg + (dy / r) * r_new;

  const float x0f = floorf(Xn), y0f = floorf(Yn);
  const float wx1 = Xn - x0f, wx0 = 1.0f - wx1;
  const float wy1 = Yn - y0f, wy0 = 1.0f - wy1;
  const int x0 = (int)x0f, y0 = (int)y0f;
  const int x1 = x0 + 1,  y1 = y0 + 1;
  const bool vx0 = (x0 >= 0) & (x0 <= WW - 1);
  const bool vx1 = (x1 >= 0) & (x1 <= WW - 1);
  const bool vy0 = (y0 >= 0) & (y0 <= HH - 1);
  const bool vy1 = (y1 >= 0) & (y1 <= HH - 1);
  const int x0c = min(max(x0, 0), WW - 1), x1c = min(max(x1, 0), WW - 1);
  const int y0c = min(max(y0, 0), HH - 1), y1c = min(max(y1, 0), HH - 1);
  const float w00 = wy0 * wx0 * (float)(vy0 & vx0);
  const float w01 = wy0 * wx1 * (float)(vy0 & vx1);
  const float w10 = wy1 * wx0 * (float)(vy1 & vx0);
  const float w11 = wy1 * wx1 * (float)(vy1 & vx1);
  const int i00 = y0c * WW + x0c, i01 = y0c * WW + x1c;
  const int i10 = y1c * WW + x0c, i11 = y1c * WW + x1c;

  const float* imb = img + (size_t)b * 3 * HWSZ;
  float* ob = out + (size_t)b * 3 * HWSZ + p;
  #pragma unroll
  for (int c = 0; c < 3; ++c) {
    const float* ch = imb + c * HWSZ;
    float v = ch[i00] * w00 + ch[i01] * w01 + ch[i10] * w10 + ch[i11] * w11;
    v = fminf(fmaxf(v, 0.0f), 255.0f);
    ob[c * HWSZ] = (float)(int)v;   // uint8 truncation semantics
  }
}

// ---------------------------------------------------------------------------
extern "C" void kernel_launch(void* const* d_in, const int* in_sizes, int n_in,
                              void* d_out, int out_size, void* d_ws, size_t ws_size,
                              hipStream_t stream) {
  const float* img  = (const float*)d_in[0];   // (32,3,512,512)
  const float* gaze = (const float*)d_in[1];   // (32,2)
  const float* sal  = (const float*)d_in[2];   // (32,1,512,512)
  float* ws = (float*)d_ws;
  float* partials = ws;                          // 32*32*3 floats
  float* params   = ws + BATCH * RB_TILES * 3;   // 32*4 floats
  float* out = (float*)d_out;

  cm_reduce_kernel<<<dim3(RB_TILES, BATCH), 256, 0, stream>>>(sal, gaze, partials);
  cm_param_kernel<<<1, 32, 0, stream>>>(partials, gaze, params);
  cm_warp_kernel<<<dim3(HWSZ / 256, BATCH), 256, 0, stream>>>(img, params, out);
}